// USTSSMBlock_67680094650382
// MI455X (gfx1250) — compile-verified
//
#include <hip/hip_runtime.h>
#include <math.h>

#define DIM      512
#define D_STATE  16
#define D_CONV   4
#define D_INNER  1024
#define DT_RANK  32
#define BB       2
#define LL       2048
#define BL       (BB*LL)   /* 4096 rows */

typedef float v2f __attribute__((ext_vector_type(2)));
typedef float v8f __attribute__((ext_vector_type(8)));

// ---------------------------------------------------------------------------
// Kernel 1: LayerNorm over last dim (512). One block (256 threads) per row.
// ---------------------------------------------------------------------------
__global__ void ln_kernel(const float* __restrict__ x, const float* __restrict__ g,
                          const float* __restrict__ b, float* __restrict__ h) {
    const int row = blockIdx.x;          // 0..BL-1
    const int tid = threadIdx.x;         // 0..255, 2 elements each
    const float2 v = reinterpret_cast<const float2*>(x + (size_t)row * DIM)[tid];
    __shared__ float ssum[256];
    __shared__ float ssq[256];
    ssum[tid] = v.x + v.y;
    ssq[tid]  = v.x * v.x + v.y * v.y;
    __syncthreads();
    for (int off = 128; off > 0; off >>= 1) {
        if (tid < off) { ssum[tid] += ssum[tid + off]; ssq[tid] += ssq[tid + off]; }
        __syncthreads();
    }
    const float mu   = ssum[0] * (1.0f / DIM);
    const float var  = ssq[0] * (1.0f / DIM) - mu * mu;
    const float rstd = rsqrtf(var + 1e-5f);
    const int c = tid * 2;
    float2 o;
    o.x = (v.x - mu) * rstd * g[c]     + b[c];
    o.y = (v.y - mu) * rstd * g[c + 1] + b[c + 1];
    reinterpret_cast<float2*>(h + (size_t)row * DIM)[tid] = o;
}

// ---------------------------------------------------------------------------
// Kernel 2: fp32 WMMA GEMM  C[M,N] = A[M,K] * W[N,K]^T  (both K-major).
// Block = 8 waves = 128x64 C tile (wave w -> m-tile w, all waves share one
// 64-row W strip). W strip staged in LDS in 32-wide K chunks, double-buffered:
// global b128 loads for chunk c+1 fly while chunk c's 32 WMMAs execute.
// f32 16x16x4 fragments: lanes 0-15 hold K=0/1, lanes 16-31 hold K=2/3,
// rows/cols indexed by lane%16 (ISA 7.12.2).
// EPI: 0 = none, 1 = softplus(c + bias[col]), 2 = c + res[row*N + col]
// ---------------------------------------------------------------------------
#define KT       32            /* K-chunk (8 wmma k-steps)            */
#define WSTRIDE  34            /* padded LDS row stride (bank-spread) */

template<int EPI>
__device__ __forceinline__ void store_tile(const v8f acc, float* __restrict__ C,
                                           int ldc, int N, int rbase, int col,
                                           const float* __restrict__ aux) {
    #pragma unroll
    for (int r = 0; r < 8; ++r) {
        const int row = rbase + r;
        float v = acc[r];
        if (EPI == 1) {
            v += aux[col];
            v = (v > 20.0f) ? v : log1pf(expf(v));     // softplus
        } else if (EPI == 2) {
            v += aux[(size_t)row * N + col];           // residual
        }
        C[(size_t)row * ldc + col] = v;
    }
}

template<int EPI>
__global__ void gemm_wmma_f32(const float* __restrict__ A, int lda,
                              const float* __restrict__ W, int ldw,
                              float* __restrict__ C, int ldc,
                              int M, int N, int K,
                              const float* __restrict__ aux) {
    __shared__ float wlds[2][64 * WSTRIDE];
    const int tid    = threadIdx.x;
    const int lane   = tid & 31;
    const int w      = tid >> 5;                 // wave 0..7
    const int mblocks= M >> 7;
    const int mblk   = blockIdx.x % mblocks;
    const int nstrip = blockIdx.x / mblocks;
    const int m0     = (mblk << 7) + (w << 4);
    const int n0     = nstrip << 6;
    const int lm     = lane & 15;
    const int khalf  = (lane >> 4) << 1;         // 0 (lanes 0-15) or 2 (16-31)

    const float* arow = A + (size_t)(m0 + lm) * lda + khalf;

    // cooperative W staging: 64 rows x 32 floats = 512 float4; thread tid
    // loads float4 #tid (rows 0-31) and #tid+256 (rows 32-63)
    const int wr = tid >> 3;                     // 0..31
    const int wc = (tid & 7) << 2;               // 0,4,..,28

    v8f acc0 = {}, acc1 = {}, acc2 = {}, acc3 = {};
    v2f a_cur[8], a_nxt[8];
    float4 wf0, wf1;

    auto gload_w = [&](int kc) {
        wf0 = *reinterpret_cast<const float4*>(W + (size_t)(n0 + wr)      * ldw + kc + wc);
        wf1 = *reinterpret_cast<const float4*>(W + (size_t)(n0 + wr + 32) * ldw + kc + wc);
    };
    auto store_w = [&](int buf) {
        float* p0 = &wlds[buf][wr * WSTRIDE + wc];
        p0[0] = wf0.x; p0[1] = wf0.y; p0[2] = wf0.z; p0[3] = wf0.w;
        float* p1 = &wlds[buf][(wr + 32) * WSTRIDE + wc];
        p1[0] = wf1.x; p1[1] = wf1.y; p1[2] = wf1.z; p1[3] = wf1.w;
    };
    auto gload_a = [&](int kc, v2f* dst) {
        #pragma unroll
        for (int s = 0; s < 8; ++s)
            dst[s] = *reinterpret_cast<const v2f*>(arow + kc + s * 4);
    };
    auto compute = [&](int buf, const v2f* areg) {
        const float* wl = &wlds[buf][0];
        #pragma unroll
        for (int s = 0; s < 8; ++s) {
            const int kk = s * 4 + khalf;
            const v2f b0 = *reinterpret_cast<const v2f*>(wl + (lm)      * WSTRIDE + kk);
            const v2f b1 = *reinterpret_cast<const v2f*>(wl + (lm + 16) * WSTRIDE + kk);
            const v2f b2 = *reinterpret_cast<const v2f*>(wl + (lm + 32) * WSTRIDE + kk);
            const v2f b3 = *reinterpret_cast<const v2f*>(wl + (lm + 48) * WSTRIDE + kk);
            acc0 = __builtin_amdgcn_wmma_f32_16x16x4_f32(false, areg[s], false, b0, (short)0, acc0, false, false);
            acc1 = __builtin_amdgcn_wmma_f32_16x16x4_f32(false, areg[s], false, b1, (short)0, acc1, false, false);
            acc2 = __builtin_amdgcn_wmma_f32_16x16x4_f32(false, areg[s], false, b2, (short)0, acc2, false, false);
            acc3 = __builtin_amdgcn_wmma_f32_16x16x4_f32(false, areg[s], false, b3, (short)0, acc3, false, false);
        }
    };

    // prologue: stage chunk 0
    gload_w(0);
    gload_a(0, a_cur);
    store_w(0);
    __syncthreads();

    const int nch = K / KT;
    for (int c = 0; c < nch; ++c) {
        const int nxt = c + 1;
        if (nxt < nch) {                 // launch next chunk's global traffic
            gload_w(nxt * KT);
            gload_a(nxt * KT, a_nxt);
        }
        compute(c & 1, a_cur);           // 32 WMMAs while loads are in flight
        if (nxt < nch) {
            store_w(nxt & 1);            // loadcnt wait lands here, post-compute
            #pragma unroll
            for (int s = 0; s < 8; ++s) a_cur[s] = a_nxt[s];
        }
        __syncthreads();
    }

    const int rbase = m0 + ((lane >> 4) << 3);
    store_tile<EPI>(acc0, C, ldc, N, rbase, n0 +  0 + lm, aux);
    store_tile<EPI>(acc1, C, ldc, N, rbase, n0 + 16 + lm, aux);
    store_tile<EPI>(acc2, C, ldc, N, rbase, n0 + 32 + lm, aux);
    store_tile<EPI>(acc3, C, ldc, N, rbase, n0 + 48 + lm, aux);
}

// ---------------------------------------------------------------------------
// Kernel 3: depthwise causal conv (width 4) + bias + SiLU.
// xp lives in xz[b,l,0:D_INNER] (row stride 2*D_INNER).
// ---------------------------------------------------------------------------
__global__ void conv_silu_kernel(const float* __restrict__ xz,
                                 const float* __restrict__ cw,
                                 const float* __restrict__ cb,
                                 float* __restrict__ xc) {
    const int idx = blockIdx.x * blockDim.x + threadIdx.x;  // BL*D_INNER threads
    const int d  = idx & (D_INNER - 1);
    const int bt = idx >> 10;                               // b*LL + t
    const int t  = bt & (LL - 1);
    float acc = cb[d];
    const float* w = cw + d * D_CONV;
    #pragma unroll
    for (int k = 0; k < D_CONV; ++k) {
        const int lag = (D_CONV - 1) - k;                   // xp_pad[k+t] == xp[t-lag]
        if (t >= lag)
            acc = fmaf(xz[(size_t)(bt - lag) * (2 * D_INNER) + d], w[k], acc);
    }
    const float s = acc / (1.0f + __expf(-acc));            // SiLU
    xc[(size_t)bt * D_INNER + d] = s;
}

// ---------------------------------------------------------------------------
// Kernel 4: selective scan. One thread per (b,d); 16 states in registers.
// B/C (proj[...,32:64], shared across all d) staged in LDS per 128-step chunk.
// Fuses  y = (scan + xc*Dp) * silu(z)  and writes y in-place over dt
// (element read strictly before write, same thread).
// ---------------------------------------------------------------------------
#define CHUNK 128
__global__ void scan_kernel(const float* __restrict__ proj,   // [BL,64]
                            const float* dt,                  // [BL,D_INNER] softplus'ed
                            const float* __restrict__ xc,     // [BL,D_INNER]
                            const float* __restrict__ xz,     // z at +D_INNER
                            const float* __restrict__ A_log,  // [D_INNER,16]
                            const float* __restrict__ Dp,     // [D_INNER]
                            float* y) {                       // aliases dt
    __shared__ float bc[CHUNK * 32];
    const int tid = threadIdx.x;
    const int b   = blockIdx.x / (D_INNER / 256);
    const int d   = (blockIdx.x % (D_INNER / 256)) * 256 + tid;

    float An[D_STATE], s[D_STATE];
    #pragma unroll
    for (int n = 0; n < D_STATE; ++n) {
        An[n] = -__expf(A_log[d * D_STATE + n]);
        s[n]  = 0.0f;
    }
    const float dpv = Dp[d];

    for (int t0 = 0; t0 < LL; t0 += CHUNK) {
        __syncthreads();
        for (int i = tid; i < CHUNK * 32; i += 256) {
            const int tt = i >> 5, c = i & 31;
            bc[i] = proj[(size_t)(b * LL + t0 + tt) * 64 + 32 + c];
        }
        __syncthreads();
        for (int ti = 0; ti < CHUNK; ++ti) {
            const size_t bt  = (size_t)b * LL + t0 + ti;
            const float dtv  = dt[bt * D_INNER + d];
            const float xcv  = xc[bt * D_INNER + d];
            const float zv   = xz[bt * (2 * D_INNER) + D_INNER + d];
            const float* bcp = bc + ti * 32;
            float yv = 0.0f;
            #pragma unroll
            for (int n = 0; n < D_STATE; ++n) {
                const float dA = __expf(dtv * An[n]);
                s[n] = fmaf(s[n], dA, dtv * bcp[n] * xcv);  // s*dA + dt*B*x
                yv   = fmaf(s[n], bcp[16 + n], yv);         // y += s*C
            }
            yv = fmaf(xcv, dpv, yv);                        // + xc*Dp
            const float sig = 1.0f / (1.0f + __expf(-zv));
            y[bt * D_INNER + d] = yv * (zv * sig);          // * silu(z)
        }
    }
}

// ---------------------------------------------------------------------------
// Launch: LN -> GEMM(xz) -> conv+SiLU -> GEMM(proj) -> GEMM(dt,softplus)
//         -> scan -> GEMM(out, +residual)
// ---------------------------------------------------------------------------
extern "C" void kernel_launch(void* const* d_in, const int* in_sizes, int n_in,
                              void* d_out, int out_size, void* d_ws, size_t ws_size,
                              hipStream_t stream) {
    (void)in_sizes; (void)n_in; (void)out_size; (void)ws_size;
    const float* x      = (const float*)d_in[0];
    const float* ln_g   = (const float*)d_in[1];
    const float* ln_b   = (const float*)d_in[2];
    const float* W_in   = (const float*)d_in[3];
    const float* conv_w = (const float*)d_in[4];
    const float* conv_b = (const float*)d_in[5];
    const float* W_xp   = (const float*)d_in[6];
    const float* W_dt   = (const float*)d_in[7];
    const float* b_dt   = (const float*)d_in[8];
    const float* A_log  = (const float*)d_in[9];
    const float* Dp     = (const float*)d_in[10];
    const float* W_out  = (const float*)d_in[11];
    float* out = (float*)d_out;

    // workspace layout (floats), ~76.5 MB total
    float* ws   = (float*)d_ws;
    float* h    = ws;                                   // BL*DIM
    float* xz   = h    + (size_t)BL * DIM;              // BL*2*D_INNER
    float* xc   = xz   + (size_t)BL * 2 * D_INNER;      // BL*D_INNER
    float* proj = xc   + (size_t)BL * D_INNER;          // BL*64
    float* dt   = proj + (size_t)BL * 64;               // BL*D_INNER (y in-place)

    ln_kernel<<<BL, 256, 0, stream>>>(x, ln_g, ln_b, h);

    {   // xz = h * W_in^T : M=4096, N=2048, K=512
        const int M = BL, N = 2 * D_INNER, K = DIM;
        const int grid = (M / 128) * (N / 64);
        gemm_wmma_f32<0><<<grid, 256, 0, stream>>>(h, DIM, W_in, DIM, xz, N, M, N, K, nullptr);
    }

    conv_silu_kernel<<<(BL * D_INNER) / 256, 256, 0, stream>>>(xz, conv_w, conv_b, xc);

    {   // proj = xc * W_xproj^T : M=4096, N=64, K=1024
        const int M = BL, N = DT_RANK + 2 * D_STATE, K = D_INNER;
        const int grid = (M / 128) * (N / 64);
        gemm_wmma_f32<0><<<grid, 256, 0, stream>>>(xc, D_INNER, W_xp, D_INNER, proj, N, M, N, K, nullptr);
    }

    {   // dt = softplus(proj[:, :32] * W_dt^T + b_dt) : M=4096, N=1024, K=32
        const int M = BL, N = D_INNER, K = DT_RANK;
        const int grid = (M / 128) * (N / 64);
        gemm_wmma_f32<1><<<grid, 256, 0, stream>>>(proj, 64, W_dt, DT_RANK, dt, N, M, N, K, b_dt);
    }

    scan_kernel<<<BB * (D_INNER / 256), 256, 0, stream>>>(proj, dt, xc, xz, A_log, Dp, dt);

    {   // out = x + y * W_out^T : M=4096, N=512, K=1024
        const int M = BL, N = DIM, K = D_INNER;
        const int grid = (M / 128) * (N / 64);
        gemm_wmma_f32<2><<<grid, 256, 0, stream>>>(dt, D_INNER, W_out, D_INNER, out, N, M, N, K, x);
    }
}